// SimplifiedMambaBlock_38835094290480
// MI455X (gfx1250) — compile-verified
//
#include <hip/hip_runtime.h>
#include <hip/hip_bf16.h>

// ---------------------------------------------------------------------------
// Mamba block for MI455X (gfx1250, wave32).
// GEMMs use V_WMMA_F32_16X16X4_F32 (fp32 matrix pipe, exact precision).
// ---------------------------------------------------------------------------

typedef float v2f __attribute__((ext_vector_type(2)));
typedef float v8f __attribute__((ext_vector_type(8)));

#define D_MODEL 1024
#define D_STATE 16
#define D_CONV  4
#define D_INNER 2048
#define BATCH   4
#define SEQ     2048
#define MROWS   (BATCH * SEQ)     // 8192

__device__ __forceinline__ float silu_f(float v) {
    return v / (1.0f + expf(-v));
}

// ---------------------------------------------------------------------------
// Kernel 1: LayerNorm. One wave32 per row (1024 elems -> 32 per lane).
// ---------------------------------------------------------------------------
__global__ __launch_bounds__(256) void ln_kernel(
    const float* __restrict__ x, const float* __restrict__ w,
    const float* __restrict__ b, float* __restrict__ xn)
{
    const int lane = threadIdx.x & 31;
    const int row  = blockIdx.x * 8 + (threadIdx.x >> 5);
    const float* xr = x + (size_t)row * D_MODEL;

    float s = 0.f, sq = 0.f;
    #pragma unroll
    for (int j = 0; j < 32; ++j) {
        const float v = xr[lane + j * 32];
        s += v; sq += v * v;
    }
    #pragma unroll
    for (int m = 16; m >= 1; m >>= 1) {
        s  += __shfl_xor(s,  m, 32);
        sq += __shfl_xor(sq, m, 32);
    }
    const float mean = s * (1.0f / D_MODEL);
    const float var  = sq * (1.0f / D_MODEL) - mean * mean;
    const float rstd = rsqrtf(var + 1e-5f);

    float* outr = xn + (size_t)row * D_MODEL;
    #pragma unroll
    for (int j = 0; j < 32; ++j) {
        const int c = lane + j * 32;
        outr[c] = (xr[c] - mean) * rstd * w[c] + b[c];
    }
}

// ---------------------------------------------------------------------------
// Kernel 2/6: WMMA fp32 GEMM.  C[M x N] = A[M x K] * Bw[K x N] + bias (+resid)
// Block = 256 threads = 8 waves; block tile 128x128; K-step 32 through LDS.
// Each wave: 32(M) x 64(N) = 2x4 tiles of v_wmma_f32_16x16x4_f32.
//
// LDS layouts (fragment-native, so the inner loop is pure ds_load_b64+wmma):
//   As[r][k]              row-major, pad 4        -> A frag = b64 at (r, kb)
//   Bs[k/2][2n + (k&1)]   K-pair interleaved      -> B frag = b64 at (k/2, 2n)
// ---------------------------------------------------------------------------
__global__ __launch_bounds__(256) void gemm128_wmma_f32(
    const float* __restrict__ A,  int lda,
    const float* __restrict__ Bw, int ldb,
    const float* __restrict__ bias,
    const float* __restrict__ resid,
    float* __restrict__ C, int ldc, int K)
{
    __shared__ __align__(16) float As[128][36];   // 18.0 KB
    __shared__ __align__(16) float Bs[16][260];   // 16.3 KB (256 data + pad 4)

    const int tid  = threadIdx.x;
    const int lane = tid & 31;
    const int w    = tid >> 5;
    const int wm   = w & 3;        // M quadrant: rows wm*32
    const int wn   = w >> 2;       // N half:     cols wn*64
    const int l16  = lane & 15;
    const int hi   = lane >> 4;    // 0 or 1 (lane half)
    const int rowBase = blockIdx.y * 128;
    const int colBase = blockIdx.x * 128;

    v8f acc[2][4];
    #pragma unroll
    for (int i = 0; i < 2; ++i)
        #pragma unroll
        for (int j = 0; j < 4; ++j)
            #pragma unroll
            for (int r = 0; r < 8; ++r) acc[i][j][r] = 0.f;

    for (int k0 = 0; k0 < K; k0 += 32) {
        // ---- Stage A tile: 128 rows x 32 k (float4 loads) -----------------
        for (int i = tid; i < 1024; i += 256) {
            const int r = i >> 3, c = (i & 7) << 2;
            const float4 v = *reinterpret_cast<const float4*>(
                &A[(size_t)(rowBase + r) * lda + k0 + c]);
            *reinterpret_cast<float4*>(&As[r][c]) = v;
        }
        // ---- Stage B tile: 32 k x 128 n as 2x2 micro-tiles ----------------
        // Bs[p][4q..4q+3] = { B[2p][2q], B[2p+1][2q], B[2p][2q+1], B[2p+1][2q+1] }
        for (int i = tid; i < 1024; i += 256) {
            const int p = i >> 6, q = i & 63;
            const float* g = &Bw[(size_t)(k0 + 2 * p) * ldb + colBase + 2 * q];
            const v2f g0 = *reinterpret_cast<const v2f*>(g);
            const v2f g1 = *reinterpret_cast<const v2f*>(g + ldb);
            float4 st; st.x = g0.x; st.y = g1.x; st.z = g0.y; st.w = g1.y;
            *reinterpret_cast<float4*>(&Bs[p][q << 2]) = st;
        }
        // ---- Prefetch next K-panel (global_prefetch_b8) -------------------
        const int kn = k0 + 32;
        if (kn < K) {
            if (tid < 128) {
                __builtin_prefetch(&A[(size_t)(rowBase + tid) * lda + kn], 0, 1);
            } else {
                const int t2 = tid - 128;
                __builtin_prefetch(
                    &Bw[(size_t)(kn + (t2 >> 2)) * ldb + colBase + ((t2 & 3) << 5)],
                    0, 1);
            }
        }
        __syncthreads();

        #pragma unroll
        for (int kk = 0; kk < 32; kk += 4) {
            const int kb = kk + (hi << 1);     // lanes 0-15: K,K+1; 16-31: K+2,K+3
            // A 16x4 fragment layout (ISA 7.12.2): v0={K0,K2}, v1={K1,K3}
            const v2f a0 = *reinterpret_cast<const v2f*>(&As[wm * 32 + l16][kb]);
            const v2f a1 = *reinterpret_cast<const v2f*>(&As[wm * 32 + 16 + l16][kb]);
            #pragma unroll
            for (int j = 0; j < 4; ++j) {
                const int c = wn * 64 + j * 16 + l16;
                // B 4x16 fragment: v0={K0,K2}? -> per-lane pair (kb, kb+1)
                const v2f bf = *reinterpret_cast<const v2f*>(&Bs[kb >> 1][c << 1]);
                acc[0][j] = __builtin_amdgcn_wmma_f32_16x16x4_f32(
                    false, a0, false, bf, (short)0, acc[0][j], false, false);
                acc[1][j] = __builtin_amdgcn_wmma_f32_16x16x4_f32(
                    false, a1, false, bf, (short)0, acc[1][j], false, false);
            }
        }
        __syncthreads();
    }

    // Epilogue: C/D layout VGPR r -> row r (lanes 0-15) / r+8 (lanes 16-31)
    #pragma unroll
    for (int i = 0; i < 2; ++i)
        #pragma unroll
        for (int j = 0; j < 4; ++j)
            #pragma unroll
            for (int r = 0; r < 8; ++r) {
                const int row = rowBase + wm * 32 + i * 16 + r + (hi << 3);
                const int col = colBase + wn * 64 + j * 16 + l16;
                float v = acc[i][j][r] + bias[col];
                if (resid) v += resid[(size_t)row * ldc + col];
                C[(size_t)row * ldc + col] = v;
            }
}

// ---------------------------------------------------------------------------
// Kernel 3: causal depthwise conv1d (taps=4, left pad 3) + SiLU.
// ---------------------------------------------------------------------------
__global__ __launch_bounds__(256) void conv_silu_kernel(
    const float* __restrict__ xg,       // (M, 4096); x_main = cols [0,2048)
    const float* __restrict__ cw,       // (2048, 1, 4)
    const float* __restrict__ cb,       // (2048)
    float* __restrict__ xc)             // (M, 2048)
{
    const int idx = blockIdx.x * 256 + threadIdx.x;   // (b*S + t)*DI + i
    const int i = idx & (D_INNER - 1);
    const int t = (idx / D_INNER) & (SEQ - 1);
    const int b = idx / (D_INNER * SEQ);

    float acc = cb[i];
    #pragma unroll
    for (int j = 0; j < D_CONV; ++j) {
        const int ts = t - (D_CONV - 1) + j;
        if (ts >= 0)
            acc += xg[(size_t)(b * SEQ + ts) * (2 * D_INNER) + i] * cw[i * D_CONV + j];
    }
    xc[idx] = silu_f(acc);
}

// ---------------------------------------------------------------------------
// Kernel 4: ssm = x_conv @ W_x + b_x   (M x 32, K = 2048). Block per row.
// ---------------------------------------------------------------------------
__global__ __launch_bounds__(256) void ssm_proj_kernel(
    const float* __restrict__ xc, const float* __restrict__ Wx,
    const float* __restrict__ bx, float* __restrict__ ssm)
{
    const int row   = blockIdx.x;
    const int n     = threadIdx.x & 31;
    const int chunk = threadIdx.x >> 5;
    const float* xr = xc + (size_t)row * D_INNER;

    float p = 0.f;
    const int k0 = chunk * 256;
    for (int k = k0; k < k0 + 256; ++k)
        p += xr[k] * Wx[(size_t)k * 32 + n];

    __shared__ float red[8][32];
    red[chunk][n] = p;
    __syncthreads();
    if (threadIdx.x < 32) {
        float s = bx[threadIdx.x];
        #pragma unroll
        for (int c = 0; c < 8; ++c) s += red[c][threadIdx.x];
        ssm[(size_t)row * 32 + threadIdx.x] = s;
    }
}

// ---------------------------------------------------------------------------
// Kernel 5: selective scan + D skip + SiLU gating.
// One lane per (batch, channel); h[16] in registers; B/C tiled through LDS.
// ---------------------------------------------------------------------------
__global__ __launch_bounds__(256) void scan_kernel(
    const float* __restrict__ xc,     // (M, 2048)
    const float* __restrict__ xg,     // gate = cols [2048,4096) of (M,4096)
    const float* __restrict__ ssm,    // (M, 32)  [B | C]
    const float* __restrict__ A_log,  // (2048, 16)
    const float* __restrict__ Dv,     // (2048)
    float* __restrict__ yp)           // (M, 2048)
{
    const int tid = threadIdx.x;
    const int b   = blockIdx.x >> 3;
    const int i   = ((blockIdx.x & 7) << 8) + tid;

    float dA[D_STATE], h[D_STATE];
    #pragma unroll
    for (int n = 0; n < D_STATE; ++n) {
        dA[n] = expf(-expf(A_log[i * D_STATE + n]));
        h[n]  = 0.f;
    }
    const float Dp = Dv[i];

    __shared__ float sB[64][D_STATE];
    __shared__ float sC[64][D_STATE];

    for (int t0 = 0; t0 < SEQ; t0 += 64) {
        const int rowBase = b * SEQ + t0;
        for (int j = tid; j < 64 * 32; j += 256) {
            const int r = j >> 5, c = j & 31;
            const float v = ssm[(size_t)(rowBase + r) * 32 + c];
            if (c < D_STATE) sB[r][c] = v; else sC[r][c - D_STATE] = v;
        }
        __syncthreads();

        for (int tt = 0; tt < 64; ++tt) {
            const size_t row = (size_t)(rowBase + tt);
            const float xv = xc[row * D_INNER + i];
            const float gv = xg[row * (2 * D_INNER) + D_INNER + i];
            float y = 0.f;
            #pragma unroll
            for (int n = 0; n < D_STATE; ++n) {
                h[n] = dA[n] * h[n] + xv * sB[tt][n];
                y += h[n] * sC[tt][n];
            }
            y += xv * Dp;
            y *= silu_f(gv);
            yp[row * D_INNER + i] = y;
        }
        __syncthreads();
    }
}

// ---------------------------------------------------------------------------
extern "C" void kernel_launch(void* const* d_in, const int* in_sizes, int n_in,
                              void* d_out, int out_size, void* d_ws, size_t ws_size,
                              hipStream_t stream) {
    (void)in_sizes; (void)n_in; (void)out_size; (void)ws_size;

    const float* x      = (const float*)d_in[0];
    const float* ln_w   = (const float*)d_in[1];
    const float* ln_b   = (const float*)d_in[2];
    const float* W_in   = (const float*)d_in[3];
    const float* b_in   = (const float*)d_in[4];
    const float* conv_w = (const float*)d_in[5];
    const float* conv_b = (const float*)d_in[6];
    const float* A_log  = (const float*)d_in[7];
    const float* Dv     = (const float*)d_in[8];
    const float* W_x    = (const float*)d_in[9];
    const float* b_x    = (const float*)d_in[10];
    const float* W_out  = (const float*)d_in[11];
    const float* b_out  = (const float*)d_in[12];
    float*       out    = (float*)d_out;

    // Workspace layout (floats)
    float* ws  = (float*)d_ws;
    float* xn  = ws;                                  // 8192*1024
    float* xg  = xn  + (size_t)MROWS * D_MODEL;       // 8192*4096
    float* xc  = xg  + (size_t)MROWS * 2 * D_INNER;   // 8192*2048
    float* ssm = xc  + (size_t)MROWS * D_INNER;       // 8192*32
    float* yp  = ssm + (size_t)MROWS * 32;            // 8192*2048

    // 1. LayerNorm (8 rows per block, wave32 per row)
    ln_kernel<<<MROWS / 8, 256, 0, stream>>>(x, ln_w, ln_b, xn);

    // 2. xg = xn @ W_in + b_in   (8192 x 4096, K=1024)
    {
        dim3 g(2 * D_INNER / 128, MROWS / 128);
        gemm128_wmma_f32<<<g, 256, 0, stream>>>(
            xn, D_MODEL, W_in, 2 * D_INNER, b_in, nullptr, xg, 2 * D_INNER, D_MODEL);
    }

    // 3. depthwise causal conv + SiLU
    conv_silu_kernel<<<(size_t)MROWS * D_INNER / 256, 256, 0, stream>>>(
        xg, conv_w, conv_b, xc);

    // 4. ssm = x_conv @ W_x + b_x   (8192 x 32, K=2048)
    ssm_proj_kernel<<<MROWS, 256, 0, stream>>>(xc, W_x, b_x, ssm);

    // 5. selective scan + D skip + gating
    scan_kernel<<<BATCH * (D_INNER / 256), 256, 0, stream>>>(
        xc, xg, ssm, A_log, Dv, yp);

    // 6. out = yp @ W_out + b_out + residual   (8192 x 1024, K=2048)
    {
        dim3 g(D_MODEL / 128, MROWS / 128);
        gemm128_wmma_f32<<<g, 256, 0, stream>>>(
            yp, D_INNER, W_out, D_MODEL, b_out, x, out, D_MODEL, D_INNER);
    }
}